// MiAMix_49452253446220
// MI455X (gfx1250) — compile-verified
//
#include <hip/hip_runtime.h>
#include <stdint.h>

// ---------------------------------------------------------------------------
// MiAMix on MI455X (gfx1250, wave32)
//   A1: parallel Marsaglia-Tsang gamma proposals (threefry2x32, JAX-style)
//   A2: first-accept scan + per-sample params (idx, modes, rects, lam_total)
//       + builds 128x128 fp32 selection matrix for the label GEMM
//   B : streaming blend out_X = m*X + (1-m)*X[idx]  (b128 loads, NT b128 stores)
//   C : out_y = A x onehot(targets) via v_wmma_f32_16x16x4_f32
// ---------------------------------------------------------------------------

#define B_N 128
#define C_N 3
#define H_N 224
#define W_N 224
#define KMAX 2
#define GITERS 64
#define NCLS 1000

typedef float f32x4 __attribute__((ext_vector_type(4)));
typedef float v2f   __attribute__((ext_vector_type(2)));
typedef float v8f   __attribute__((ext_vector_type(8)));

struct JKey { uint32_t a, b; };

__device__ __forceinline__ uint32_t rotl32(uint32_t v, int d) {
    return (v << d) | (v >> (32 - d));
}

// Threefry-2x32, 20 rounds (JAX PRNG core)
__device__ __forceinline__ void tf2x32(uint32_t k0, uint32_t k1,
                                       uint32_t& x0, uint32_t& x1) {
    const uint32_t ks2 = k0 ^ k1 ^ 0x1BD11BDAu;
    uint32_t ks[3] = {k0, k1, ks2};
    const int ra[4] = {13, 15, 26, 6};
    const int rb[4] = {17, 29, 16, 24};
    x0 += k0; x1 += k1;
    #pragma unroll
    for (int g = 0; g < 5; ++g) {
        const int* r = (g & 1) ? rb : ra;
        #pragma unroll
        for (int i = 0; i < 4; ++i) {
            x0 += x1; x1 = rotl32(x1, r[i]); x1 ^= x0;
        }
        x0 += ks[(g + 1) % 3];
        x1 += ks[(g + 2) % 3] + (uint32_t)(g + 1);
    }
}

// Element i of jax random_bits(key, 32, (n,)): counters split in halves.
__device__ __forceinline__ uint32_t jax_bits(JKey k, uint32_t n, uint32_t i) {
    uint32_t half = (n + 1u) >> 1;
    bool second = (i >= half);
    uint32_t lane = second ? (i - half) : i;
    uint32_t x0 = lane;
    uint32_t x1v = half + lane;
    uint32_t x1 = (x1v < n) ? x1v : 0u;   // zero pad for odd n
    tf2x32(k.a, k.b, x0, x1);
    return second ? x1 : x0;
}

// jax.random.split(key, m)[j]  (reshape(m,2) of 2m counter bits)
__device__ __forceinline__ JKey jax_split(JKey k, uint32_t m, uint32_t j) {
    JKey r;
    r.a = jax_bits(k, 2u * m, 2u * j);
    r.b = jax_bits(k, 2u * m, 2u * j + 1u);
    return r;
}

__device__ __forceinline__ float jax_uniform(JKey k, uint32_t n, uint32_t i,
                                             float lo, float hi) {
    uint32_t b = jax_bits(k, n, i);
    float f = __uint_as_float((b >> 9) | 0x3f800000u) - 1.0f;  // [0,1)
    return fmaxf(lo, f * (hi - lo) + lo);
}

// single-precision erfinv (M. Giles approximation)
__device__ __forceinline__ float erfinv_f(float x) {
    float w = -logf((1.0f - x) * (1.0f + x));
    float p;
    if (w < 5.0f) {
        w -= 2.5f;
        p = 2.81022636e-08f;
        p = fmaf(p, w, 3.43273939e-07f);
        p = fmaf(p, w, -3.5233877e-06f);
        p = fmaf(p, w, -4.39150654e-06f);
        p = fmaf(p, w, 0.00021858087f);
        p = fmaf(p, w, -0.00125372503f);
        p = fmaf(p, w, -0.00417768164f);
        p = fmaf(p, w, 0.246640727f);
        p = fmaf(p, w, 1.50140941f);
    } else {
        w = sqrtf(w) - 3.0f;
        p = -0.000200214257f;
        p = fmaf(p, w, 0.000100950558f);
        p = fmaf(p, w, 0.00134934322f);
        p = fmaf(p, w, -0.00367342844f);
        p = fmaf(p, w, 0.00573950773f);
        p = fmaf(p, w, -0.0076224613f);
        p = fmaf(p, w, 0.00943887047f);
        p = fmaf(p, w, 1.00167406f);
        p = fmaf(p, w, 2.83297682f);
    }
    return p * x;
}

__device__ __forceinline__ float jax_normal(JKey k, uint32_t n, uint32_t i) {
    float u = jax_uniform(k, n, i, -0.99999994f, 1.0f);
    return 1.41421356f * erfinv_f(u);
}

__device__ __forceinline__ int imin(int a, int b) { return a < b ? a : b; }
__device__ __forceinline__ int imax(int a, int b) { return a > b ? a : b; }

// ---------------------------------------------------------------------------
// Kernel A1: all (b, j, t) gamma proposals in parallel.  i = b*3+j in [0,384)
// ---------------------------------------------------------------------------
__global__ void __launch_bounds__(256)
miamix_rng_iter(float* __restrict__ cand, int* __restrict__ acc) {
    int gid = blockIdx.x * 256 + threadIdx.x;     // 24576 total
    int t = gid & 63;
    int i = gid >> 6;                             // 0..383
    int b = i / 3;
    int j = i - 3 * b;

    JKey root = {0u, 42u};                        // jax.random.key(42)
    JKey K3 = jax_split(root, 7, 3);
    JKey ks = jax_split(K3, 2, 1);
    JKey kt = jax_split(ks, GITERS, (uint32_t)t);
    JKey k1 = jax_split(kt, 2, 0);
    JKey k2 = jax_split(kt, 2, 1);

    float x = jax_normal(k1, 384, (uint32_t)i);
    float u = jax_uniform(k2, 384, (uint32_t)i, 1e-12f, 1.0f);

    float a = 1.0f;
    if (j == 2) {
        JKey Kk = jax_split(root, 7, 2);
        a = 1.0f + (float)(jax_bits(Kk, B_N, (uint32_t)b) & 1u);  // kvals*ALPHA
    }
    a = fmaxf(a, 1e-8f);
    float ab = (a > 1.0f) ? a : a + 1.0f;
    float d = ab - (1.0f / 3.0f);
    float c = 1.0f / sqrtf(9.0f * d);

    float v0 = 1.0f + c * x;
    float v = v0 * v0 * v0;
    float logv = logf(fmaxf(v, 1e-30f));
    float x2 = x * x;
    bool acpt = (v > 0.0f) &&
                ((u < 1.0f - 0.0331f * x2 * x2) ||
                 (logf(u) < 0.5f * x2 + d * (1.0f - v + logv)));

    cand[i * GITERS + t] = d * v;
    acc[i * GITERS + t] = acpt ? 1 : 0;
}

// ---------------------------------------------------------------------------
// Kernel A2: first-accept scan, boost, per-sample params + selection matrix
// ---------------------------------------------------------------------------
__global__ void __launch_bounds__(384)
miamix_rng_final(const float* __restrict__ cand, const int* __restrict__ acc,
                 int* __restrict__ sidx, float* __restrict__ slamtot,
                 int* __restrict__ smode, float* __restrict__ slam,
                 int* __restrict__ srect, float* __restrict__ Amat) {
    __shared__ float gam_s[384];
    int tid = threadIdx.x;
    JKey root = {0u, 42u};

    // zero the 128x128 selection matrix cooperatively
    for (int q = tid; q < B_N * B_N; q += 384) Amat[q] = 0.0f;

    {   // per-(b,j) gamma result
        int i = tid;
        int b = i / 3;
        int j = i - 3 * b;
        float a = 1.0f;
        if (j == 2) {
            JKey Kk = jax_split(root, 7, 2);
            a = 1.0f + (float)(jax_bits(Kk, B_N, (uint32_t)b) & 1u);
        }
        a = fmaxf(a, 1e-8f);
        float ab = (a > 1.0f) ? a : a + 1.0f;
        float d = ab - (1.0f / 3.0f);
        float out = d;
        bool done = false;
        for (int t = 0; t < GITERS; ++t) {
            int ac = acc[i * GITERS + t];
            float cv = cand[i * GITERS + t];
            if (ac && !done) { out = cv; done = true; }
        }
        JKey K3 = jax_split(root, 7, 3);
        JKey kb = jax_split(K3, 2, 0);
        float U = jax_uniform(kb, 384, (uint32_t)i, 1e-12f, 1.0f);
        float boost = (a > 1.0f) ? 1.0f : powf(U, 1.0f / a);
        gam_s[i] = out * boost;
    }
    __syncthreads();

    if (tid < B_N) {
        int b = tid;
        float g0 = gam_s[3 * b], g1 = gam_s[3 * b + 1], g2 = gam_s[3 * b + 2];
        float s = g0 + g1 + g2;
        float lamv[2] = {g0 / s, g1 / s};

        JKey K0 = jax_split(root, 7, 0);
        JKey K1 = jax_split(root, 7, 1);
        JKey Kk = jax_split(root, 7, 2);
        JKey K4 = jax_split(root, 7, 4);
        JKey K5 = jax_split(root, 7, 5);
        JKey K6 = jax_split(root, 7, 6);

        uint32_t ridx = jax_bits(K0, B_N, (uint32_t)b) % (uint32_t)B_N;
        bool selfm = jax_uniform(K1, B_N, (uint32_t)b, 0.0f, 1.0f) < 0.1f;
        int idx = selfm ? b : (int)ridx;
        int kval = 1 + (int)(jax_bits(Kk, B_N, (uint32_t)b) & 1u);

        int mode[2], rx1[2], rx2[2], ry1[2], ry2[2];
        for (int k = 0; k < KMAX; ++k) {
            uint32_t e = (uint32_t)(2 * b + k);
            int meth = (jax_uniform(K4, 256, e, 0.0f, 1.0f) >=
                        (float)(2.0 / 3.0)) ? 1 : 0;
            float cutv = sqrtf(fminf(fmaxf(1.0f - lamv[k], 0.0f), 1.0f));
            int cw = (int)floorf((float)W_N * cutv);
            int ch = (int)floorf((float)H_N * cutv);
            int cx = (int)(jax_bits(K5, 256, e) % (uint32_t)W_N);
            int cy = (int)(jax_bits(K6, 256, e) % (uint32_t)H_N);
            rx1[k] = imax(cx - cw / 2, 0);
            rx2[k] = imin(cx + cw / 2, W_N);
            ry1[k] = imax(cy - ch / 2, 0);
            ry2[k] = imin(cy + ch / 2, H_N);
            mode[k] = (k < kval) ? meth : 2;   // 0=mix, 1=cut, 2=inactive
        }

        // closed-form lam_total = mean over HxW of the mask product
        const float HW = (float)(H_N * W_N);
        float area0 = (float)(imax(rx2[0] - rx1[0], 0) * imax(ry2[0] - ry1[0], 0));
        float area1 = (float)(imax(rx2[1] - rx1[1], 0) * imax(ry2[1] - ry1[1], 0));
        float mean;
        if (mode[0] == 2 && mode[1] == 2)      mean = 1.0f;
        else if (mode[0] == 2)                 mean = (mode[1] == 0) ? lamv[1] : area1 / HW;
        else if (mode[1] == 2)                 mean = (mode[0] == 0) ? lamv[0] : area0 / HW;
        else if (mode[0] == 0 && mode[1] == 0) mean = lamv[0] * lamv[1];
        else if (mode[0] == 0)                 mean = lamv[0] * (area1 / HW);
        else if (mode[1] == 0)                 mean = lamv[1] * (area0 / HW);
        else {
            int iw = imax(imin(rx2[0], rx2[1]) - imax(rx1[0], rx1[1]), 0);
            int ih = imax(imin(ry2[0], ry2[1]) - imax(ry1[0], ry1[1]), 0);
            mean = (float)(iw * ih) / HW;
        }
        float lt = fminf(fmaxf(mean, 0.0f), 1.0f);

        sidx[b] = idx;
        slamtot[b] = lt;
        for (int k = 0; k < KMAX; ++k) {
            smode[k * B_N + b] = mode[k];
            slam[k * B_N + b] = lamv[k];
            int* r = srect + 4 * (k * B_N + b);
            r[0] = rx1[k]; r[1] = rx2[k]; r[2] = ry1[k]; r[3] = ry2[k];
        }
        float* row = Amat + (size_t)b * B_N;
        row[b] = lt;
        row[idx] += 1.0f - lt;   // idx==b collapses to exactly lt+(1-lt)
    }
}

// ---------------------------------------------------------------------------
// Kernel B: out_X = m*X + (1-m)*X[idx], mask from per-sample params.
// grid (49, 3, 128) x 256 threads; one float4 per thread; NT stores.
// ---------------------------------------------------------------------------
__global__ void __launch_bounds__(256)
miamix_blend(const float* __restrict__ img, float* __restrict__ outX,
             const int* __restrict__ sidx, const int* __restrict__ smode,
             const float* __restrict__ slam, const int* __restrict__ srect) {
    const int b = blockIdx.z, c = blockIdx.y;
    const int p = blockIdx.x * 256 + threadIdx.x;   // 0..12543 (224*56)
    const int h = p / 56;
    const int wq = p - h * 56;

    const int ib = sidx[b];
    const int m0 = smode[b], m1 = smode[B_N + b];
    const float l0 = slam[b], l1 = slam[B_N + b];
    const int* r0 = srect + 4 * b;
    const int* r1 = srect + 4 * (B_N + b);
    const bool row0 = (h >= r0[2]) && (h < r0[3]);
    const bool row1 = (h >= r1[2]) && (h < r1[3]);

    const size_t self = (((size_t)b * C_N + c) * H_N + h) * 56 + wq;
    const size_t part = (((size_t)ib * C_N + c) * H_N + h) * 56 + wq;
    const f32x4* imgv = (const f32x4*)img;
    f32x4 x = imgv[self];
    f32x4 xt = imgv[part];

    f32x4 o;
    const int wbase = wq * 4;
    #pragma unroll
    for (int e = 0; e < 4; ++e) {
        int w = wbase + e;
        float t0 = (m0 == 0) ? l0
                 : (m0 == 1) ? ((row0 && w >= r0[0] && w < r0[1]) ? 1.0f : 0.0f)
                 : 1.0f;
        float t1 = (m1 == 0) ? l1
                 : (m1 == 1) ? ((row1 && w >= r1[0] && w < r1[1]) ? 1.0f : 0.0f)
                 : 1.0f;
        float m = t0 * t1;
        o[e] = m * x[e] + (1.0f - m) * xt[e];
    }
    __builtin_nontemporal_store(o, ((f32x4*)outX) + self);
}

// ---------------------------------------------------------------------------
// Kernel C: out_y = A(128x128) x onehot(targets)(128x1000) via fp32 WMMA.
// One wave per 16x16 D tile: block = 8 waves (M blocks), grid.x = 63 N blocks.
// A layout (ISA 7.12.2, 32-bit A 16x4): lanes0-15 VGPR{0,1}=K{0,1};
// lanes16-31=K{2,3}. B (4x16) assumed mirrored: lanes0-15 hold K{0,1},
// lanes16-31 hold K{2,3}, N = lane&15.
// ---------------------------------------------------------------------------
#if defined(__has_builtin)
#if __has_builtin(__builtin_amdgcn_wmma_f32_16x16x4_f32)
#define USE_WMMA_F32 1
#endif
#endif

#ifdef USE_WMMA_F32
__global__ void __launch_bounds__(256)
miamix_labels_wmma(const float* __restrict__ Amat,
                   const int* __restrict__ targets,
                   float* __restrict__ outY) {
    const int lane = threadIdx.x & 31;
    const int mblk = threadIdx.x >> 5;   // 0..7 -> rows [mblk*16, +16)
    const int nblk = blockIdx.x;         // 0..62 -> cols [nblk*16, +16)
    const int l = lane & 15;
    const bool hi2 = lane >= 16;
    const int n = nblk * 16 + l;
    const int row = mblk * 16 + l;

    v8f cacc = {};
    for (int k0 = 0; k0 < B_N; k0 += 4) {
        const int kb = k0 + (hi2 ? 2 : 0);
        v2f av, bv;
        av.x = Amat[(size_t)row * B_N + kb];
        av.y = Amat[(size_t)row * B_N + kb + 1];
        const int t0 = targets[kb];
        const int t1 = targets[kb + 1];
        bv.x = (t0 == n) ? 1.0f : 0.0f;
        bv.y = (t1 == n) ? 1.0f : 0.0f;
        cacc = __builtin_amdgcn_wmma_f32_16x16x4_f32(
            false, av, false, bv, (short)0, cacc, false, false);
    }
    if (n < NCLS) {
        #pragma unroll
        for (int r = 0; r < 8; ++r) {
            int m = mblk * 16 + r + (hi2 ? 8 : 0);
            outY[(size_t)m * NCLS + n] = cacc[r];
        }
    }
}
#else
__global__ void __launch_bounds__(256)
miamix_labels_plain(const int* __restrict__ targets,
                    const int* __restrict__ sidx,
                    const float* __restrict__ slamtot,
                    float* __restrict__ outY) {
    int e = blockIdx.x * 256 + threadIdx.x;
    if (e >= B_N * NCLS) return;
    int b = e / NCLS, n = e - NCLS * b;
    float lt = slamtot[b];
    int ib = sidx[b];
    float y  = (targets[b]  == n) ? 1.0f : 0.0f;
    float yt = (targets[ib] == n) ? 1.0f : 0.0f;
    outY[e] = lt * y + (1.0f - lt) * yt;
}
#endif

// ---------------------------------------------------------------------------
extern "C" void kernel_launch(void* const* d_in, const int* in_sizes, int n_in,
                              void* d_out, int out_size, void* d_ws,
                              size_t ws_size, hipStream_t stream) {
    const float* images = (const float*)d_in[0];
    const int* targets = (const int*)d_in[1];
    float* outX = (float*)d_out;
    float* outY = outX + (size_t)B_N * C_N * H_N * W_N;

    char* ws = (char*)d_ws;
    float* cand    = (float*)(ws + 0);        // 384*64 f32  =  98304 B
    int*   acc     = (int*)  (ws + 98304);    // 384*64 i32  =  98304 B
    int*   sidx    = (int*)  (ws + 196608);   // 128 i32
    float* slamtot = (float*)(ws + 197120);   // 128 f32
    int*   smode   = (int*)  (ws + 197632);   // 256 i32
    float* slam    = (float*)(ws + 198656);   // 256 f32
    int*   srect   = (int*)  (ws + 199680);   // 256*4 i32   =   4096 B
    float* Amat    = (float*)(ws + 204032);   // 128*128 f32 =  65536 B

    // A1: 24576 proposal threads
    miamix_rng_iter<<<96, 256, 0, stream>>>(cand, acc);
    // A2: single block finalizes all per-sample parameters
    miamix_rng_final<<<1, 384, 0, stream>>>(cand, acc, sidx, slamtot,
                                            smode, slam, srect, Amat);
    // B: streaming blend, 128*3*224*56 float4 elements
    miamix_blend<<<dim3(49, 3, 128), 256, 0, stream>>>(images, outX, sidx,
                                                       smode, slam, srect);
    // C: label mixing
#ifdef USE_WMMA_F32
    miamix_labels_wmma<<<63, 256, 0, stream>>>(Amat, targets, outY);
#else
    miamix_labels_plain<<<(B_N * NCLS + 255) / 256, 256, 0, stream>>>(
        targets, sidx, slamtot, outY);
#endif
}